// HausdorffDistance_77386720740128
// MI455X (gfx1250) — compile-verified
//
#include <hip/hip_runtime.h>
#include <hip/hip_bf16.h>
#include <math.h>

// Hausdorff distance via WMMA F32 16x16x4: dist^2 expressed as a K=4 bilinear
// form  [X, Y, |o'|^2, 1] . [-2x, -2y, 1, |p|^2]  with EPS folded into the
// outer point (reference: diff = outer - inner + eps).
//
// v2: inner points are packed ONCE per LDS page into WMMA B-operand form, so
// the hot loop is just ds_load_b64 -> v_wmma -> v_min3 (no per-tile VALU
// construction). Raw staging still uses CDNA5 global_load_async_to_lds.

typedef __attribute__((ext_vector_type(2))) float v2f;
typedef __attribute__((ext_vector_type(8))) float v8f;

#define HD_EPS 1e-6f
#define PAGE 2048          // inner points per LDS page: 16 KB raw + 32 KB packed
#define BLOCK 512          // 16 waves (wave32)
#define TILE_OUTER 256     // 16 waves * 16 outer points per wave

__global__ __launch_bounds__(BLOCK)
void HausdorffDistance_wmma_kernel(const float* __restrict__ c1,
                                   const float* __restrict__ c2,
                                   int L1, int L2,
                                   unsigned int* __restrict__ acc /* B entries: f32 bits of max(min dist^2) */)
{
    __shared__ float2 sRaw[PAGE];        // raw inner points (async DMA target)
    __shared__ float2 sPack[2 * PAGE];   // packed B operand: [2i]=(-2x,-2y), [2i+1]=(1,|p|^2)

    const int batch = blockIdx.y;
    const int dir   = blockIdx.z;  // 0: h21 (outer=c2+eps, inner=c1), 1: h12 (outer=c1+eps, inner=c2)
    const int Lout  = dir ? L1 : L2;
    const int Lin   = dir ? L2 : L1;
    const float2* outer = (const float2*)(dir ? c1 : c2) + (size_t)batch * Lout;
    const float2* inner = (const float2*)(dir ? c2 : c1) + (size_t)batch * Lin;

    const int tid  = threadIdx.x;
    const int lane = tid & 31;
    const int wave = tid >> 5;
    const bool hi  = lane >= 16;   // lanes 16-31 hold K=2,3 of A / B per ISA 32-bit layouts

    // ---- A matrix: 16 outer points per wave, fixed for the whole kernel ----
    int m = blockIdx.x * TILE_OUTER + wave * 16 + (lane & 15);
    m = m < Lout ? m : Lout - 1;          // clamp: duplicated outer rows don't change the max
    float2 q = outer[m];
    float X = q.x + HD_EPS, Y = q.y + HD_EPS;
    v2f a;
    a.x = hi ? (X * X + Y * Y) : X;       // lanes<16: A[m][0]=X      lanes>=16: A[m][2]=|o'|^2
    a.y = hi ? 1.0f            : Y;       // lanes<16: A[m][1]=Y      lanes>=16: A[m][3]=1

    v8f vmin;
#pragma unroll
    for (int i = 0; i < 8; ++i) vmin[i] = 3.0e38f;
    const v8f czero = {};

    // Per-lane packed-B slot: byte offset = n*16 + h*8 with n = t*16 + (lane&15)
    const int laneSlot = 2 * (lane & 15) + (hi ? 1 : 0);

    const int nPages = (Lin + PAGE - 1) / PAGE;
    for (int page = 0; page < nPages; ++page) {
        // ---- stage raw inner points into LDS via CDNA5 async-to-LDS DMA ----
        for (int i = tid; i < PAGE; i += BLOCK) {
            int pt = page * PAGE + i;
            pt = pt < Lin ? pt : Lin - 1;                           // clamp: dup inner points keep min correct
            unsigned int ldsb = (unsigned int)(size_t)(&sRaw[i]);   // LDS byte offset (low 32 bits of flat addr)
            unsigned int voff = (unsigned int)pt * 8u;              // GVS: saddr64 + vaddr32
            asm volatile("global_load_async_to_lds_b64 %0, %1, %2 offset:0"
                         :: "v"(ldsb), "v"(voff), "s"(inner) : "memory");
        }
        asm volatile("s_wait_asynccnt 0x0" ::: "memory");
        // Each thread packs exactly the slots it staged -> no barrier needed here.
#pragma unroll
        for (int i = tid; i < PAGE; i += BLOCK) {
            float2 p = sRaw[i];
            sPack[2 * i]     = make_float2(-2.0f * p.x, -2.0f * p.y); // B[0][n], B[1][n]
            sPack[2 * i + 1] = make_float2(1.0f, p.x * p.x + p.y * p.y); // B[2][n], B[3][n]
        }
        __syncthreads();

        int cnt = Lin - page * PAGE; if (cnt > PAGE) cnt = PAGE;
        const int tCount = (cnt + 15) >> 4;
#pragma unroll 8
        for (int t = 0; t < tCount; ++t) {
            float2 pk = sPack[(t << 5) + laneSlot];   // one ds_load_b64, conflict-free
            v2f bb; bb.x = pk.x; bb.y = pk.y;
            // D[m][n] = dist^2(outer m, inner n) in f32
            v8f d = __builtin_amdgcn_wmma_f32_16x16x4_f32(
                false, a, false, bb, (short)0, czero, false, false);
#pragma unroll
            for (int i = 0; i < 8; ++i) vmin[i] = fminf(vmin[i], d[i]);
        }
        __syncthreads();
    }

    // ---- per-row min across the 16 columns (xor<16 stays within each half) ----
#pragma unroll
    for (int s = 1; s < 16; s <<= 1) {
#pragma unroll
        for (int i = 0; i < 8; ++i)
            vmin[i] = fminf(vmin[i], __shfl_xor(vmin[i], s, 32));
    }
    // ---- max over the wave's 16 rows ----
    float mx = vmin[0];
#pragma unroll
    for (int i = 1; i < 8; ++i) mx = fmaxf(mx, vmin[i]);
    mx = fmaxf(mx, __shfl_xor(mx, 16, 32));
    mx = fmaxf(mx, 0.0f);  // guard tiny negative from fp cancellation; keeps int-max trick valid
    if (lane == 0)
        atomicMax((int*)(acc + batch), __float_as_int(mx));
}

__global__ void HausdorffDistance_init_kernel(unsigned int* acc, int B) {
    int i = threadIdx.x + blockIdx.x * blockDim.x;
    if (i < B) acc[i] = 0u;
}

__global__ void HausdorffDistance_final_kernel(const unsigned int* __restrict__ acc,
                                               const float* __restrict__ res,
                                               float* __restrict__ out, int B) {
    int i = threadIdx.x + blockIdx.x * blockDim.x;
    if (i < B) out[i] = sqrtf(__int_as_float((int)acc[i])) * res[i];
}

extern "C" void kernel_launch(void* const* d_in, const int* in_sizes, int n_in,
                              void* d_out, int out_size, void* d_ws, size_t ws_size,
                              hipStream_t stream) {
    const float* c1  = (const float*)d_in[0];
    const float* c2  = (const float*)d_in[1];
    const float* res = (const float*)d_in[2];
    const int B  = in_sizes[2];
    const int L1 = in_sizes[0] / (B * 2);
    const int L2 = in_sizes[1] / (B * 2);
    unsigned int* acc = (unsigned int*)d_ws;

    HausdorffDistance_init_kernel<<<(B + 255) / 256, 256, 0, stream>>>(acc, B);

    const int maxL = L1 > L2 ? L1 : L2;
    const int nChunks = (maxL + TILE_OUTER - 1) / TILE_OUTER;
    dim3 grid(nChunks, B, 2);
    HausdorffDistance_wmma_kernel<<<grid, BLOCK, 0, stream>>>(c1, c2, L1, L2, acc);

    HausdorffDistance_final_kernel<<<(B + 255) / 256, 256, 0, stream>>>(acc, res, (float*)d_out, B);
}